// PlanetoidGCN_46866683134518
// MI455X (gfx1250) — compile-verified
//
#include <hip/hip_runtime.h>
#include <hip/hip_bf16.h>

typedef __attribute__((ext_vector_type(2))) float v2f;
typedef __attribute__((ext_vector_type(8))) float v8f;

// -----------------------------------------------------------------------------
// Kernel 1: h = x @ W using V_WMMA_F32_16X16X4_F32 (fp32 in, fp32 accumulate).
// Block = 256 threads = 8 wave32s arranged 2 (M) x 4 (N) -> 32x64 block tile.
// K = 128 consumed in 32 steps of K=4.
// A 16x4 layout (ISA 7.12.2): lanes 0-15 hold {K0,K1}, lanes 16-31 hold {K2,K3},
// M = lane%16 per VGPR pair. B 4x16 mirrors it (row K striped across lanes).
// C/D 16x16: VGPR r -> (M=r, N=lane) lanes 0-15 ; (M=r+8, N=lane-16) lanes 16-31.
// -----------------------------------------------------------------------------
__global__ void gcn_gemm_wmma(const float* __restrict__ x,
                              const float* __restrict__ W,
                              float* __restrict__ h) {
  const int lane  = threadIdx.x & 31;
  const int wave  = threadIdx.x >> 5;
  const int waveM = wave >> 2;          // 0..1
  const int waveN = wave & 3;           // 0..3
  const int half  = lane >> 4;          // 0: lanes 0-15, 1: lanes 16-31
  const int l16   = lane & 15;

  const int rowBase = blockIdx.x * 32 + waveM * 16;   // M tile origin
  const int colBase = waveN * 16;                     // N tile origin

  const float* __restrict__ arow = x + (size_t)(rowBase + l16) * 128;

  v8f c = {0.f, 0.f, 0.f, 0.f, 0.f, 0.f, 0.f, 0.f};

#pragma unroll 4
  for (int k0 = 0; k0 < 128; k0 += 4) {
    const int ka = k0 + 2 * half;
    v2f a, b;
    a.x = arow[ka];
    a.y = arow[ka + 1];
    b.x = W[(size_t)ka * 64 + colBase + l16];
    b.y = W[(size_t)(ka + 1) * 64 + colBase + l16];
    c = __builtin_amdgcn_wmma_f32_16x16x4_f32(
        /*neg_a=*/false, a, /*neg_b=*/false, b,
        /*c_mod=*/(short)0, c, /*reuse_a=*/false, /*reuse_b=*/false);
  }

  // Store 16x16 tile: row r of C lives in VGPR r (lanes 0-15), r+8 (lanes 16-31).
  float* __restrict__ orow = h + (size_t)rowBase * 64 + colBase + l16;
  const int mOff = half * 8;
#pragma unroll
  for (int r = 0; r < 8; ++r) {
    orow[(size_t)(r + mOff) * 64] = c[r];
  }
}

// -----------------------------------------------------------------------------
// Kernel 2: degree histogram over target nodes (float add of 1.0 is exact here).
// -----------------------------------------------------------------------------
__global__ void gcn_deg(const long long* __restrict__ col,
                        float* __restrict__ deg, long long E) {
  long long e = (long long)blockIdx.x * blockDim.x + threadIdx.x;
  if (e < E) {
    atomicAdd(&deg[col[e]], 1.0f);
  }
}

// -----------------------------------------------------------------------------
// Kernel 3: deg -> deg^{-1/2} in place (0 where deg == 0).
// -----------------------------------------------------------------------------
__global__ void gcn_inv_sqrt(float* __restrict__ deg, int N) {
  int i = blockIdx.x * blockDim.x + threadIdx.x;
  if (i < N) {
    float d = deg[i];
    deg[i] = (d > 0.0f) ? rsqrtf(fmaxf(d, 1.0f)) : 0.0f;
  }
}

// -----------------------------------------------------------------------------
// Kernel 4: edge scatter. One wave32 per edge; lane owns features {2l, 2l+1}.
// h and out both fit in the 192MB L2, so the gather + f32 atomics stay on-chip.
// -----------------------------------------------------------------------------
__global__ void gcn_scatter(const long long* __restrict__ row,
                            const long long* __restrict__ col,
                            const float* __restrict__ h,
                            const float* __restrict__ dis,
                            float* __restrict__ out, long long E) {
  const int lane = threadIdx.x & 31;
  const long long e = (long long)blockIdx.x * 8 + (threadIdx.x >> 5);
  if (e >= E) return;

  const long long r = row[e];
  const long long t = col[e];
  const float norm = dis[r] * dis[t];

  const float2 hv = *(const float2*)(h + r * 64 + lane * 2);
  float* dst = out + t * 64 + lane * 2;
  atomicAdd(dst,     hv.x * norm);
  atomicAdd(dst + 1, hv.y * norm);
}

// -----------------------------------------------------------------------------
// Kernel 5: out = PReLU(out + b), in place.
// -----------------------------------------------------------------------------
__global__ void gcn_finalize(float* __restrict__ out,
                             const float* __restrict__ b,
                             const float* __restrict__ prelu_a,
                             long long total) {
  long long i = (long long)blockIdx.x * blockDim.x + threadIdx.x;
  if (i < total) {
    float v = out[i] + b[i & 63];
    out[i] = (v >= 0.0f) ? v : prelu_a[0] * v;
  }
}

extern "C" void kernel_launch(void* const* d_in, const int* in_sizes, int n_in,
                              void* d_out, int out_size, void* d_ws, size_t ws_size,
                              hipStream_t stream) {
  const float*     x   = (const float*)d_in[0];
  const long long* ei  = (const long long*)d_in[1];   // int64 per reference
  const float*     W   = (const float*)d_in[2];
  const float*     b   = (const float*)d_in[3];
  const float*     pa  = (const float*)d_in[4];

  const int       N = in_sizes[0] / 128;      // 100000
  const long long E = (long long)in_sizes[1] / 2;  // 1600000

  const long long* row = ei;        // edge_index[0]
  const long long* col = ei + E;    // edge_index[1]

  float* out = (float*)d_out;                     // [N, 64]
  float* h   = (float*)d_ws;                      // [N, 64] scratch (25.6 MB)
  float* deg = h + (size_t)N * 64;                // [N]     scratch (400 KB)

  // Zero accumulators every call (graph-capture safe memset nodes).
  hipMemsetAsync(out, 0, (size_t)N * 64 * sizeof(float), stream);
  hipMemsetAsync(deg, 0, (size_t)N * sizeof(float), stream);

  // 1) h = x @ W  (WMMA fp32; N = 100000 = 3125 * 32, so EXEC all-1s everywhere)
  gcn_gemm_wmma<<<N / 32, 256, 0, stream>>>(x, W, h);

  // 2) degree over targets
  gcn_deg<<<(int)((E + 255) / 256), 256, 0, stream>>>(col, deg, E);

  // 3) deg^{-1/2}
  gcn_inv_sqrt<<<(N + 255) / 256, 256, 0, stream>>>(deg, N);

  // 4) gather * norm -> scatter-add (one wave per edge)
  gcn_scatter<<<(int)((E + 7) / 8), 256, 0, stream>>>(row, col, h, deg, out, E);

  // 5) bias + PReLU
  const long long total = (long long)N * 64;
  gcn_finalize<<<(int)((total + 255) / 256), 256, 0, stream>>>(out, b, pa, total);
}